// HybridKernel_65481071407372
// MI455X (gfx1250) — compile-verified
//
#include <hip/hip_runtime.h>

typedef float v2f __attribute__((ext_vector_type(2)));
typedef float v4f __attribute__((ext_vector_type(4)));
typedef float v8f __attribute__((ext_vector_type(8)));

#define GAMMA_C     1.0f
#define THRESHOLD_C 0.0f

// ---------------------------------------------------------------------------
// Wave32 cross-lane sum using V_WMMA_F32_16X16X4_F32.
// A layout (32-bit 16x4, ISA 7.12.2): lane L<16  -> A[M=L][K=0],A[M=L][K=1]
//                                     lane L>=16 -> A[M=L-16][K=2],A[M=L-16][K=3]
// With a = {s, 0} and B = all-ones, C[m][n] = s_m + s_{m+16}.
// C layout: lane L<16 holds M=0..7 (VGPR0..7), lane L>=16 holds M=8..15.
// Sum the 8 C values per lane, then one shfl_xor(16) completes the wave sum.
// ---------------------------------------------------------------------------
__device__ __forceinline__ float wave_reduce_wmma(float s) {
#if __has_builtin(__builtin_amdgcn_wmma_f32_16x16x4_f32)
  v2f a;  a[0] = s;     a[1] = 0.0f;
  v2f bv; bv[0] = 1.0f; bv[1] = 1.0f;
  v8f c = {};
  c = __builtin_amdgcn_wmma_f32_16x16x4_f32(
      /*neg_a=*/false, a, /*neg_b=*/false, bv,
      /*c_mod=*/(short)0, c, /*reuse_a=*/false, /*reuse_b=*/false);
  float h = ((c[0] + c[1]) + (c[2] + c[3])) + ((c[4] + c[5]) + (c[6] + c[7]));
  h += __shfl_xor(h, 16, 32);
  return h;
#else
  for (int m = 16; m >= 1; m >>= 1) s += __shfl_xor(s, m, 32);
  return s;
#endif
}

// ---------------------------------------------------------------------------
// sigmoid(a) - sigmoid(b) == 0.5*(tanh(a/2) - tanh(b/2)).
// gfx1250 has hardware v_tanh_f32 (1 TRANS op). We return the *unscaled*
// tanh difference; the caller folds the 1/4 factor into the final sum.
// ---------------------------------------------------------------------------
__device__ __forceinline__ float tanh_half(float t) {
#if __has_builtin(__builtin_amdgcn_tanhf)
  return __builtin_amdgcn_tanhf(0.5f * t);
#else
  // Fallback: 2*sigmoid(t)-1 with raw v_rcp_f32 (no precise-div expansion).
  float s = __builtin_amdgcn_rcpf(1.0f + __expf(-t));
  return fmaf(2.0f, s, -1.0f);
#endif
}

// ---------------------------------------------------------------------------
// Kernel 1: bandwidth-bound streaming partial reduction.
// Per lane per step: one float4 row of x and of y (global_load_b128, NT),
// FMA dot + bias, two v_tanh_f32, squared-diff accumulate.
// 2-way unrolled grid-stride loop (dual accumulators) for MLP.
// ---------------------------------------------------------------------------
__global__ void __launch_bounds__(256)
rbf_partial_kernel(const float* __restrict__ x, const float* __restrict__ y,
                   const float* __restrict__ w, const float* __restrict__ b,
                   float* __restrict__ partials, int n) {
  const v4f   wv  = *(const v4f*)w;
  const float bias = b[0] - THRESHOLD_C;

  const int stride = (int)(gridDim.x * blockDim.x);
  int i = (int)(blockIdx.x * blockDim.x + threadIdx.x);

  float acc0 = 0.0f, acc1 = 0.0f;

  for (; i + stride < n; i += 2 * stride) {
    v4f xv0 = __builtin_nontemporal_load((const v4f*)(x + (size_t)i * 4));
    v4f yv0 = __builtin_nontemporal_load((const v4f*)(y + (size_t)i * 4));
    v4f xv1 = __builtin_nontemporal_load((const v4f*)(x + (size_t)(i + stride) * 4));
    v4f yv1 = __builtin_nontemporal_load((const v4f*)(y + (size_t)(i + stride) * 4));

    float lx0 = fmaf(xv0[0], wv[0], fmaf(xv0[1], wv[1],
                fmaf(xv0[2], wv[2], fmaf(xv0[3], wv[3], bias))));
    float ly0 = fmaf(yv0[0], wv[0], fmaf(yv0[1], wv[1],
                fmaf(yv0[2], wv[2], fmaf(yv0[3], wv[3], bias))));
    float lx1 = fmaf(xv1[0], wv[0], fmaf(xv1[1], wv[1],
                fmaf(xv1[2], wv[2], fmaf(xv1[3], wv[3], bias))));
    float ly1 = fmaf(yv1[0], wv[0], fmaf(yv1[1], wv[1],
                fmaf(yv1[2], wv[2], fmaf(yv1[3], wv[3], bias))));

    float d0 = tanh_half(lx0) - tanh_half(ly0);
    float d1 = tanh_half(lx1) - tanh_half(ly1);
    acc0 = fmaf(d0, d0, acc0);
    acc1 = fmaf(d1, d1, acc1);
  }
  for (; i < n; i += stride) {
    v4f xv = __builtin_nontemporal_load((const v4f*)(x + (size_t)i * 4));
    v4f yv = __builtin_nontemporal_load((const v4f*)(y + (size_t)i * 4));
    float lx = fmaf(xv[0], wv[0], fmaf(xv[1], wv[1],
               fmaf(xv[2], wv[2], fmaf(xv[3], wv[3], bias))));
    float ly = fmaf(yv[0], wv[0], fmaf(yv[1], wv[1],
               fmaf(yv[2], wv[2], fmaf(yv[3], wv[3], bias))));
    float d = tanh_half(lx) - tanh_half(ly);
    acc0 = fmaf(d, d, acc0);
  }

  // Fold the (1/2)^2 tanh-identity scale once, then WMMA wave reduction
  // (EXEC all-ones here: the loops have reconverged).
  float wsum = wave_reduce_wmma(0.25f * (acc0 + acc1));

  __shared__ float smem[8];  // 256 threads = 8 wave32s
  const int lane = threadIdx.x & 31;
  const int wid  = threadIdx.x >> 5;
  if (lane == 0) smem[wid] = wsum;
  __syncthreads();

  if (threadIdx.x == 0) {
    float t = 0.0f;
    const int nw = blockDim.x >> 5;
    for (int k = 0; k < nw; ++k) t += smem[k];
    partials[blockIdx.x] = t;  // one deterministic write per block
  }
}

// ---------------------------------------------------------------------------
// Kernel 2: reduce block partials, finish with exp(-gamma * sum).
// ---------------------------------------------------------------------------
__global__ void __launch_bounds__(256)
rbf_final_kernel(const float* __restrict__ partials, int nb,
                 float* __restrict__ out) {
  __shared__ float smem[256];
  float s = 0.0f;
  for (int i = threadIdx.x; i < nb; i += blockDim.x) s += partials[i];
  smem[threadIdx.x] = s;
  __syncthreads();

  if (threadIdx.x < 32) {
    float t = 0.0f;
    for (int k = 0; k < 8; ++k) t += smem[threadIdx.x * 8 + k];
    float total = wave_reduce_wmma(t);
    if (threadIdx.x == 0) out[0] = __expf(-GAMMA_C * total);
  }
}

// ---------------------------------------------------------------------------
extern "C" void kernel_launch(void* const* d_in, const int* in_sizes, int n_in,
                              void* d_out, int out_size, void* d_ws, size_t ws_size,
                              hipStream_t stream) {
  const float* x = (const float*)d_in[0];
  const float* y = (const float*)d_in[1];
  const float* w = (const float*)d_in[2];
  const float* b = (const float*)d_in[3];
  float* out      = (float*)d_out;
  float* partials = (float*)d_ws;

  const int n = in_sizes[0] / 4;  // B rows, F=4 floats each

  int nb = 2048;  // 2048 blocks x 256 threads -> 32 rows/thread at B=16.7M
  if ((size_t)nb * sizeof(float) > ws_size) {
    nb = (int)(ws_size / sizeof(float));
    if (nb < 1) nb = 1;
  }

  rbf_partial_kernel<<<nb, 256, 0, stream>>>(x, y, w, b, partials, n);
  rbf_final_kernel<<<1, 256, 0, stream>>>(partials, nb, out);
}